// ConstantVelocityModel_15633680957997
// MI455X (gfx1250) — compile-verified
//
#include <hip/hip_runtime.h>

#define EPSF 1e-12f
#define NQUAD 10

typedef __attribute__((ext_vector_type(2)))  float    v2f;
typedef __attribute__((ext_vector_type(8)))  float    v8f;
typedef __attribute__((ext_vector_type(16))) _Float16 v16h;

__device__ __forceinline__ float fast_sqrtf(float x) {
  return __builtin_amdgcn_sqrtf(x);          // raw v_sqrt_f32, no fixup chain
}

__device__ __forceinline__ float wave_reduce32(float v) {
#pragma unroll
  for (int off = 16; off > 0; off >>= 1) v += __shfl_xor(v, off, 32);
  return v;
}

__global__ void k_init(double* acc) {
  if (threadIdx.x < 2) acc[threadIdx.x] = 0.0;
}

// ---------------- Event term: sum over events of (beta - dist) ---------------
__global__ void k_events(const float2* __restrict__ z0, const float2* __restrict__ v0,
                         const float* __restrict__ beta_p, const float* __restrict__ t,
                         const int2* __restrict__ uv, int ne, double* __restrict__ acc) {
  const float beta = beta_p[0];
  float local = 0.f;
  for (int e = blockIdx.x * blockDim.x + threadIdx.x; e < ne;
       e += gridDim.x * blockDim.x) {
    int2  p  = uv[e];
    float tt = t[e];
    float2 zu = z0[p.x], zv = z0[p.y];
    float2 vu = v0[p.x], vv = v0[p.y];
    float dx = fmaf(vu.x - vv.x, tt, zu.x - zv.x);
    float dy = fmaf(vu.y - vv.y, tt, zu.y - zv.y);
    local += beta - fast_sqrtf(fmaf(dx, dx, fmaf(dy, dy, EPSF)));
  }
  local = wave_reduce32(local);
  __shared__ float s[8];
  const int wid = threadIdx.x >> 5;
  if ((threadIdx.x & 31) == 0) s[wid] = local;
  __syncthreads();
  if (threadIdx.x == 0) {
    float tot = 0.f;
    const int nw = blockDim.x >> 5;
    for (int w = 0; w < nw; ++w) tot += s[w];
    atomicAdd(&acc[0], (double)tot);         // global_atomic_add_f64
  }
}

// -------- Pair term: 16x16 tiles of the strict lower triangle via WMMA -------
// dist^2(i,j) = |p_i|^2 + |p_j|^2 - 2 p_i.p_j ; Gram matrix from one
// V_WMMA_F32_16X16X4_F32 per tile per quadrature time (2-D coords padded K=4).
__global__ void __launch_bounds__(32)
k_pairs(const float2* __restrict__ z0, const float2* __restrict__ v0,
        const float* __restrict__ beta_p, const float* __restrict__ t0_p,
        const float* __restrict__ tn_p, int npts, double* __restrict__ acc) {
  const int lane = threadIdx.x;      // one wave32 per block / tile
  const int li   = lane & 15;
  const int hi   = lane >> 4;        // 0: holds K0,K1 data; 1: zero (K2,K3) lanes

  // decode triangular tile index -> (bi, bj), bi >= bj
  int T = (int)blockIdx.x;
  int bi = (int)((fast_sqrtf(8.0f * (float)T + 1.0f) - 1.0f) * 0.5f);
  while ((bi + 1) * (bi + 2) / 2 <= T) ++bi;
  while (bi * (bi + 1) / 2 > T) --bi;
  const int bj = T - bi * (bi + 1) / 2;
  const int I0 = bi * 16, J0 = bj * 16;

  const float beta = beta_p[0];
  const float t0 = t0_p[0], tn = tn_p[0];
  const float dtq = (tn - t0) / (float)NQUAD;

  const int iIdx = min(I0 + li, npts - 1);
  const int jIdx = min(J0 + li, npts - 1);
  const float2 zi = z0[iIdx], vi = v0[iIdx];
  const float2 zj = z0[jIdx], vj = v0[jIdx];

  __shared__ float sI[16], sJ[16];
  const int jglob = J0 + li;

  // q-invariant validity mask per D-matrix element (branch-free inner loop)
  float msk[8];
#pragma unroll
  for (int r = 0; r < 8; ++r) {
    const int iglob = I0 + r + (hi << 3);
    msk[r] = (iglob > jglob && iglob < npts) ? 1.f : 0.f;
  }

  float sum = 0.f;

#pragma unroll 2
  for (int q = 0; q < NQUAD; ++q) {
    const float t = fmaf((float)q + 0.5f, dtq, t0);
    float ax = 0.f, ay = 0.f, bx = 0.f, by = 0.f;
    if (hi == 0) {                             // lanes 0-15 carry real data
      ax = fmaf(vi.x, t, zi.x);
      ay = fmaf(vi.y, t, zi.y);
      bx = fmaf(vj.x, t, zj.x);
      by = fmaf(vj.y, t, zj.y);
      sI[li] = fmaf(ax, ax, ay * ay);
      sJ[li] = fmaf(bx, bx, by * by);
    }
    __syncthreads();                           // 1-wave WG: NOP + ds waits

    v2f A = {ax, ay};                          // A: M=lane row, K0..K1 (K2,K3=0)
    v2f B = {bx, by};                          // B: N=lane col, K0..K1 (K2,K3=0)
    v8f c0 = {};
    v8f g = __builtin_amdgcn_wmma_f32_16x16x4_f32(false, A, false, B,
                                                  (short)0, c0, false, false);

    // D layout: element r of lane -> (M = r + 8*hi, N = lane&15)
    const float nj = sJ[li];
    float ni[8];
#pragma unroll
    for (int r = 0; r < 8; ++r) ni[r] = sI[r + (hi << 3)];
#pragma unroll
    for (int r = 0; r < 8; ++r) {
      float d2 = fmaf(-2.0f, g[r], ni[r] + nj);
      d2 = fmaxf(d2, 0.f) + EPSF;
      const float dist = fast_sqrtf(d2);
      sum = fmaf(msk[r], __expf(beta - dist), sum);
    }
    __syncthreads();
  }

  sum = wave_reduce32(sum);
  if (lane == 0) atomicAdd(&acc[1], (double)sum);
}

__global__ void k_final(const double* __restrict__ acc, const float* __restrict__ t0_p,
                        const float* __restrict__ tn_p, float* __restrict__ out) {
  const float dtq = (tn_p[0] - t0_p[0]) / (float)NQUAD;
  out[0] = (float)(acc[0] - (double)dtq * acc[1]);   // NON_EVENT_WEIGHT == 1
}

extern "C" void kernel_launch(void* const* d_in, const int* in_sizes, int n_in,
                              void* d_out, int out_size, void* d_ws, size_t ws_size,
                              hipStream_t stream) {
  const float2* z0     = (const float2*)d_in[0];
  const float2* v0     = (const float2*)d_in[1];
  const float*  beta   = (const float*)d_in[2];
  const float*  data_t = (const float*)d_in[3];
  const float*  t0     = (const float*)d_in[4];
  const float*  tn     = (const float*)d_in[5];
  const int2*   uv     = (const int2*)d_in[6];
  // d_in[7], d_in[8] (pair_u/pair_v) are regenerated from tile coordinates.

  const int npts = in_sizes[0] / 2;
  const int ne   = in_sizes[3];
  double* acc = (double*)d_ws;
  float*  out = (float*)d_out;

  k_init<<<1, 32, 0, stream>>>(acc);

  int evBlocks = (ne + 255) / 256;
  if (evBlocks > 8192) evBlocks = 8192;
  k_events<<<evBlocks, 256, 0, stream>>>(z0, v0, beta, data_t, uv, ne, acc);

  const int tb = (npts + 15) / 16;
  const int ntiles = tb * (tb + 1) / 2;              // 8256 for N=2048
  k_pairs<<<ntiles, 32, 0, stream>>>(z0, v0, beta, t0, tn, npts, acc);

  k_final<<<1, 1, 0, stream>>>(acc, t0, tn, out);
}